// SocialConv_70892730188375
// MI455X (gfx1250) — compile-verified
//
#include <hip/hip_runtime.h>
#include <hip/hip_bf16.h>

typedef __attribute__((ext_vector_type(2))) float v2f;
typedef __attribute__((ext_vector_type(8))) float v8f;

#define D_FEAT 128

// ---------------- Phase 1: zero degrees (also used to zero the CSR pad) ----------------
__global__ void zero_int_kernel(int* __restrict__ p, int n) {
    int i = blockIdx.x * blockDim.x + threadIdx.x;
    if (i < n) p[i] = 0;
}

// ---------------- Phase 2: in-degree count ----------------
__global__ void count_deg_kernel(const int* __restrict__ edge_dst, int* __restrict__ deg, int e) {
    int i = blockIdx.x * blockDim.x + threadIdx.x;
    if (i < e) atomicAdd(&deg[edge_dst[i]], 1);
}

// ---------------- Phase 3: exclusive scan (single block, 1024 threads) ----------------
__global__ __launch_bounds__(1024)
void scan_kernel(const int* __restrict__ deg, int* __restrict__ row_ptr, int n) {
    __shared__ int sdata[1024];
    __shared__ int carry_s;
    const int tid = threadIdx.x;
    if (tid == 0) carry_s = 0;
    __syncthreads();
    for (int base = 0; base < n; base += 1024) {
        int i = base + tid;
        int v = (i < n) ? deg[i] : 0;
        sdata[tid] = v;
        __syncthreads();
        for (int off = 1; off < 1024; off <<= 1) {
            int t = (tid >= off) ? sdata[tid - off] : 0;
            __syncthreads();
            sdata[tid] += t;
            __syncthreads();
        }
        int carry = carry_s;
        if (i < n) row_ptr[i] = carry + sdata[tid] - v;   // exclusive
        __syncthreads();
        if (tid == 0) carry_s = carry + sdata[1023];
        __syncthreads();
    }
    if (threadIdx.x == 0) row_ptr[n] = carry_s;
}

// ---------------- Phase 4: CSR scatter ----------------
__global__ void init_cursor_kernel(const int* __restrict__ row_ptr, int* __restrict__ cursor, int n) {
    int i = blockIdx.x * blockDim.x + threadIdx.x;
    if (i < n) cursor[i] = row_ptr[i];
}

__global__ void scatter_kernel(const int* __restrict__ edge_src, const int* __restrict__ edge_dst,
                               int* __restrict__ cursor, int* __restrict__ csr_src, int e) {
    int i = blockIdx.x * blockDim.x + threadIdx.x;
    if (i < e) {
        int pos = atomicAdd(&cursor[edge_dst[i]], 1);
        csr_src[pos] = edge_src[i];
    }
}

// ---------------- Phase 5: pull-aggregate via f32 WMMA indicator trick ----------------
// Block = 256 threads = 8 waves. Window = 16 dst nodes. Wave w owns feature tile [16w, 16w+16).
// Per 4-edge chunk of the window's CSR range:
//   A(16x4)[m,k] = (edge k in node m's range) ? 1 : 0   (two int compares vs row_ptr)
//   B(4x16)[k,n] = emb[csr_src[edge k]][fb + n]
//   D += A x B   via V_WMMA_F32_16X16X4_F32
// csr_src is padded with 8 zero entries, so tail reads need NO clamping: any edge index
// >= end reads a valid node row and its indicator is 0 (e >= rp_m1 for all m).
__global__ __launch_bounds__(256)
void aggregate_wmma_kernel(const float* __restrict__ emb,
                           const int* __restrict__ row_ptr,
                           const int* __restrict__ csr_src,
                           float* __restrict__ out, int n_nodes) {
    const int tid  = threadIdx.x;
    const int lane = tid & 31;
    const int wave = tid >> 5;          // 0..7 -> feature tile
    const int base = blockIdx.x * 16;   // first dst node of window
    const int m    = lane & 15;         // row (A) / col (B,D) id within tile
    const int hi   = lane >> 4;         // half-wave: K pair {0,1} vs {2,3}

    __shared__ int s_rp[17];
    __shared__ __align__(16) float s_inv[16];
    if (tid < 17) {
        int idx = base + tid;
        if (idx > n_nodes) idx = n_nodes;
        s_rp[tid] = row_ptr[idx];
    }
    __syncthreads();
    if (tid < 16) {
        int cnt = s_rp[tid + 1] - s_rp[tid];
        s_inv[tid] = (cnt > 0) ? (1.0f / (float)cnt) : 0.0f;   // 16 real divides per block
    }
    __syncthreads();

    const int rp_m  = s_rp[m];
    const int rp_m1 = s_rp[m + 1];
    // Uniform loop bounds -> scalar loop control; EXEC stays all-1s for WMMA.
    const int start = __builtin_amdgcn_readfirstlane(s_rp[0]);
    const int end   = __builtin_amdgcn_readfirstlane(s_rp[16]);
    const unsigned col = (unsigned)(wave * 16 + m);   // feature column this lane owns

    v8f acc = {};   // C/D accumulator, 16x16 f32 tile

    for (int e = start; e < end; e += 8) {
        // This lane supplies K = 2*hi, 2*hi+1 of each 4-edge chunk; two chunks per iter.
        int e0 = e + (hi << 1);
        int s0 = csr_src[(unsigned)e0];
        int s1 = csr_src[(unsigned)e0 + 1u];
        int s2 = csr_src[(unsigned)e0 + 4u];
        int s3 = csr_src[(unsigned)e0 + 5u];

        float b0 = emb[(unsigned)s0 * 128u + col];
        float b1 = emb[(unsigned)s1 * 128u + col];
        float b2 = emb[(unsigned)s2 * 128u + col];
        float b3 = emb[(unsigned)s3 * 128u + col];

        v2f A0 = { (e0     >= rp_m && e0     < rp_m1) ? 1.0f : 0.0f,
                   (e0 + 1 >= rp_m && e0 + 1 < rp_m1) ? 1.0f : 0.0f };
        v2f B0 = { b0, b1 };
        acc = __builtin_amdgcn_wmma_f32_16x16x4_f32(false, A0, false, B0, (short)0, acc, false, false);

        v2f A1 = { (e0 + 4 >= rp_m && e0 + 4 < rp_m1) ? 1.0f : 0.0f,
                   (e0 + 5 >= rp_m && e0 + 5 < rp_m1) ? 1.0f : 0.0f };
        v2f B1 = { b2, b3 };
        acc = __builtin_amdgcn_wmma_f32_16x16x4_f32(false, A1, false, B1, (short)0, acc, false, false);
    }

    // Fetch this half-wave's 8 reciprocals with two 16B LDS reads.
    float inv[8];
    {
        const float4 i0 = *(const float4*)&s_inv[hi << 3];
        const float4 i1 = *(const float4*)&s_inv[(hi << 3) + 4];
        inv[0] = i0.x; inv[1] = i0.y; inv[2] = i0.z; inv[3] = i0.w;
        inv[4] = i1.x; inv[5] = i1.y; inv[6] = i1.z; inv[7] = i1.w;
    }

    // D layout: VGPR j holds rows M=j (lanes 0-15) and M=8+j (lanes 16-31), N = m.
    if (base + 16 <= n_nodes) {            // uniform (SGPR) branch: full-window fast path
        #pragma unroll
        for (int j = 0; j < 8; ++j) {
            int row = j + (hi << 3);
            out[(unsigned)(base + row) * 128u + col] = acc[j] * inv[j];
        }
    } else {
        #pragma unroll
        for (int j = 0; j < 8; ++j) {
            int row = j + (hi << 3);
            if (base + row < n_nodes)
                out[(unsigned)(base + row) * 128u + col] = acc[j] * inv[j];
        }
    }
}

extern "C" void kernel_launch(void* const* d_in, const int* in_sizes, int n_in,
                              void* d_out, int out_size, void* d_ws, size_t ws_size,
                              hipStream_t stream) {
    const float* emb = (const float*)d_in[0];
    const int*  esrc = (const int*)d_in[1];
    const int*  edst = (const int*)d_in[2];
    float* out = (float*)d_out;

    const int E = in_sizes[1];
    const int N = in_sizes[0] / D_FEAT;
    const int PAD = 8;   // csr_src tail pad so the aggregate loop never clamps

    // Workspace layout (256B-aligned): deg[N], row_ptr[N+1], cursor[N], csr_src[E+PAD]
    char* ws = (char*)d_ws;
    size_t off = 0;
    auto take = [&](size_t bytes) { size_t r = off; off = (off + bytes + 255) & ~(size_t)255; return r; };
    int* deg     = (int*)(ws + take((size_t)N * 4));
    int* row_ptr = (int*)(ws + take((size_t)(N + 1) * 4));
    int* cursor  = (int*)(ws + take((size_t)N * 4));
    int* csr_src = (int*)(ws + take((size_t)(E + PAD) * 4));
    (void)ws_size;

    const int T = 256;
    zero_int_kernel   <<<(N + T - 1) / T, T, 0, stream>>>(deg, N);
    zero_int_kernel   <<<1, PAD, 0, stream>>>(csr_src + E, PAD);   // pad -> node 0 (valid row)
    count_deg_kernel  <<<(E + T - 1) / T, T, 0, stream>>>(edst, deg, E);
    scan_kernel       <<<1, 1024, 0, stream>>>(deg, row_ptr, N);
    init_cursor_kernel<<<(N + T - 1) / T, T, 0, stream>>>(row_ptr, cursor, N);
    scatter_kernel    <<<(E + T - 1) / T, T, 0, stream>>>(esrc, edst, cursor, csr_src, E);

    const int nWindows = (N + 15) / 16;
    aggregate_wmma_kernel<<<nWindows, 256, 0, stream>>>(emb, row_ptr, csr_src, out, N);
}